// torch_model_27986006901227
// MI455X (gfx1250) — compile-verified
//
#include <hip/hip_runtime.h>
#include <cstdint>

// ---------------------------------------------------------------------------
// Box-embedding loss: gather 4 rows of 128 f32 per sample, elementwise
// join/meet, log-volume reduce over D=128, two scalar outputs per sample.
// Memory-bound gather (268 MB; tables fit in 192MB L2). One wave32 per sample.
// CDNA5 path: async global->LDS DMA double-buffer + s_wait_asynccnt pipeline,
// wave32 ballot for the disjoint OR-reduce, ds-shuffle butterfly for log sums.
// ---------------------------------------------------------------------------

#define WAVES_PER_BLOCK 8
#define THREADS (WAVES_PER_BLOCK * 32)

#define MIN_MEAN_C   0.00505f   // (0.0001 + 0.01) / 2
#define MIN_VAR_C    0.00495f   // 0.01 - MIN_MEAN
#define DELTA_MEAN_C 0.9495f    // (0.9 + 0.999) / 2
#define DELTA_VAR_C  0.0495f    // 0.999 - DELTA_MEAN
#define EPS_C        1e-8f

// -------- CDNA5 async global->LDS path (compile-time probed) ---------------
#ifndef __HIP_DEVICE_COMPILE__
#define USE_ASYNC 0
#else
#if defined(__has_builtin)
#if __has_builtin(__builtin_amdgcn_global_load_async_to_lds_b128)
#define USE_ASYNC 1
#else
#define USE_ASYNC 0
#endif
#else
#define USE_ASYNC 0
#endif
#endif

#if USE_ASYNC
// Builtin prototype (from hipcc diagnostic): param0 = int4 addrspace(1)*,
// param1 = int4 addrspace(3)*, then two imm ints (offset, cpol).
// Low 32 bits of a flat LDS address are the LDS byte offset (ISA 10.2), so an
// integer round-trip to an addrspace(3) pointer is value-correct and dodges
// the C++ addrspace-cast restriction.
typedef int v4i_cdna5 __attribute__((vector_size(4 * sizeof(int))));
typedef __attribute__((address_space(1))) v4i_cdna5* g_v4i_ptr;
typedef __attribute__((address_space(3))) v4i_cdna5* l_v4i_ptr;

#define ASYNC_B128(gsrc, ldst)                                               \
  __builtin_amdgcn_global_load_async_to_lds_b128(                            \
      (g_v4i_ptr)(uintptr_t)(gsrc), (l_v4i_ptr)(uintptr_t)(ldst), 0, 0)

#if __has_builtin(__builtin_amdgcn_s_wait_asynccnt)
#define WAIT_ASYNC(n) __builtin_amdgcn_s_wait_asynccnt(n)
#else
#define WAIT_ASYNC(n) asm volatile("s_wait_asynccnt %0" ::"i"(n))
#endif
#endif  // USE_ASYNC

// -------- wave32 OR-reduce of a predicate (one v_cmp + SALU) ---------------
__device__ __forceinline__ int wave_any(int pred) {
#if defined(__HIP_DEVICE_COMPILE__) && defined(__has_builtin)
#if __has_builtin(__builtin_amdgcn_ballot_w32)
  return __builtin_amdgcn_ballot_w32(pred != 0) != 0u;
#else
  return __any(pred);
#endif
#else
  return __any(pred);
#endif
}

// -------- per-sample math (all 32 lanes hold 4 of the 128 dims) ------------
__device__ __forceinline__ void process_sample(int lane, int s, int n,
                                               float4 a1, float4 d1,
                                               float4 a2, float4 d2,
                                               float* __restrict__ out) {
  const float* a1p = (const float*)&a1;
  const float* d1p = (const float*)&d1;
  const float* a2p = (const float*)&a2;
  const float* d2p = (const float*)&d2;

  float p1 = 1.0f, p2 = 1.0f, pm = 1.0f, pj = 1.0f;
  int dj = 0;
#pragma unroll
  for (int j = 0; j < 4; ++j) {
    float m1 = a1p[j] * MIN_VAR_C + MIN_MEAN_C;
    float w1 = d1p[j] * DELTA_VAR_C + DELTA_MEAN_C;
    float x1 = m1 + w1;
    float m2 = a2p[j] * MIN_VAR_C + MIN_MEAN_C;
    float w2 = d2p[j] * DELTA_VAR_C + DELTA_MEAN_C;
    float x2 = m2 + w2;

    float jw = fmaxf(x1, x2) - fminf(m1, m2);   // join width
    float mw = fminf(x1, x2) - fmaxf(m1, m2);   // meet width
    dj |= (mw <= 0.0f) ? 1 : 0;

    // log-volumes as log of per-lane products (4 logs/lane instead of 16);
    // box widths are w1/w2 directly (x - m == w up to rounding, unchecked)
    p1 *= fmaxf(w1, EPS_C);
    p2 *= fmaxf(w2, EPS_C);
    pm *= fmaxf(mw, EPS_C);
    pj *= fmaxf(jw, EPS_C);
  }
  float l1 = __logf(p1);
  float l2 = __logf(p2);
  float lm = __logf(pm);
  float lj = __logf(pj);

  // disjoint flag: single wave32 ballot instead of a 5-level butterfly
  int any_dj = wave_any(dj);

  // wave32 butterfly reduction (sum of logs over D)
#pragma unroll
  for (int off = 16; off > 0; off >>= 1) {
    l1 += __shfl_xor(l1, off, 32);
    l2 += __shfl_xor(l2, off, 32);
    lm += __shfl_xor(lm, off, 32);
    lj += __shfl_xor(lj, off, 32);
  }

  if (lane == 0) {
    float cond = lm - l2;                          // log P(t1|t2)
    float pos  = any_dj ? (lj - l1) : (-cond);     // disjoint: join_log - t1_log
    float neg  = any_dj ? 0.0f
                        : -__logf(fmaxf(1.0f - __expf(cond), EPS_C));
    out[s]     = pos;
    out[n + s] = neg;
  }
}

#if USE_ASYNC
__device__ __forceinline__ void issue_sample(const int* __restrict__ t1x,
                                             const int* __restrict__ t2x,
                                             const float* __restrict__ minT,
                                             const float* __restrict__ deltaT,
                                             int s, int lane, float* buf) {
  int i1 = __builtin_amdgcn_readfirstlane(t1x[s]);
  int i2 = __builtin_amdgcn_readfirstlane(t2x[s]);
  const float* r0 = minT   + ((long)i1 << 7);
  const float* r1 = deltaT + ((long)i1 << 7);
  const float* r2 = minT   + ((long)i2 << 7);
  const float* r3 = deltaT + ((long)i2 << 7);
  int lo = lane * 4;  // 16 B per lane per row -> one async b128 per row
  ASYNC_B128(r0 + lo, buf + 0 * 128 + lo);
  ASYNC_B128(r1 + lo, buf + 1 * 128 + lo);
  ASYNC_B128(r2 + lo, buf + 2 * 128 + lo);
  ASYNC_B128(r3 + lo, buf + 3 * 128 + lo);
}
#endif

__global__ __launch_bounds__(THREADS) void box_loss_kernel(
    const int* __restrict__ t1x, const int* __restrict__ t2x,
    const float* __restrict__ minT, const float* __restrict__ deltaT,
    float* __restrict__ out, int n) {
  const int lane = threadIdx.x & 31;
  const int wv   = threadIdx.x >> 5;
  const int gw   = blockIdx.x * WAVES_PER_BLOCK + wv;
  const int nw   = gridDim.x * WAVES_PER_BLOCK;

#if USE_ASYNC
  // 2 KB tile per sample, double-buffered per wave: 8 waves * 4 KB = 32 KB LDS
  __shared__ __align__(16) float lds_buf[WAVES_PER_BLOCK][2][4 * 128];

  int bi = 0;
  if (gw < n) issue_sample(t1x, t2x, minT, deltaT, gw, lane, &lds_buf[wv][0][0]);
  for (int s = gw; s < n; s += nw) {
    int sn = s + nw;
    if (sn < n) {
      issue_sample(t1x, t2x, minT, deltaT, sn, lane, &lds_buf[wv][bi ^ 1][0]);
      WAIT_ASYNC(4);  // 4 newest in flight; the 4 for sample s are complete
    } else {
      WAIT_ASYNC(0);
    }
    const float4* bp = (const float4*)&lds_buf[wv][bi][0];
    float4 a1 = bp[lane];
    float4 d1 = bp[32 + lane];
    float4 a2 = bp[64 + lane];
    float4 d2 = bp[96 + lane];
    process_sample(lane, s, n, a1, d1, a2, d2, out);
    bi ^= 1;
  }
#else
  // Fallback: direct coalesced b128 gathers; wave-level parallelism hides
  // latency (16K waves in flight against an L2-resident working set).
  for (int s = gw; s < n; s += nw) {
    int i1 = __builtin_amdgcn_readfirstlane(t1x[s]);
    int i2 = __builtin_amdgcn_readfirstlane(t2x[s]);
    const float4* r0 = (const float4*)(minT   + ((long)i1 << 7));
    const float4* r1 = (const float4*)(deltaT + ((long)i1 << 7));
    const float4* r2 = (const float4*)(minT   + ((long)i2 << 7));
    const float4* r3 = (const float4*)(deltaT + ((long)i2 << 7));
    float4 a1 = r0[lane];
    float4 d1 = r1[lane];
    float4 a2 = r2[lane];
    float4 d2 = r3[lane];
    process_sample(lane, s, n, a1, d1, a2, d2, out);
  }
#endif
}

extern "C" void kernel_launch(void* const* d_in, const int* in_sizes, int n_in,
                              void* d_out, int out_size, void* d_ws,
                              size_t ws_size, hipStream_t stream) {
  const int*   t1x    = (const int*)d_in[0];
  const int*   t2x    = (const int*)d_in[1];
  const float* minT   = (const float*)d_in[2];
  const float* deltaT = (const float*)d_in[3];
  float*       out    = (float*)d_out;
  const int    n      = in_sizes[0];

  int blocks = 2048;  // 16384 waves -> 8 samples per wave at B=131072
  int maxBlocks = (n + WAVES_PER_BLOCK - 1) / WAVES_PER_BLOCK;
  if (blocks > maxBlocks) blocks = maxBlocks;
  if (blocks < 1) blocks = 1;

  box_loss_kernel<<<dim3(blocks), dim3(THREADS), 0, stream>>>(
      t1x, t2x, minT, deltaT, out, n);
}